// QuantumLayer_58025008169533
// MI455X (gfx1250) — compile-verified
//
#include <hip/hip_runtime.h>
#include <math.h>

// ---------------------------------------------------------------------------
// 12-qubit state-vector simulator, one circuit per workgroup.
// State: 4096 complex64 amplitudes = 32 KB LDS.
// Fused RZ*RY*RZ -> one 2x2 complex unitary -> real 4x4 matrix applied to
// 2048 (re0,im0,re1,im1) 4-vectors via V_WMMA_F32_16X16X4_F32 (full fp32).
// CNOT blocks are composed into a single GF(2)-linear index permutation.
// ---------------------------------------------------------------------------

typedef __attribute__((ext_vector_type(2))) float v2f;
typedef __attribute__((ext_vector_type(8))) float v8f;

#define NQ       12
#define NSTATES  4096
#define NPAIRS   2048
#define NLAY     3
#define THREADS  128              // 4 waves (wave32)
#define NWAVES   (THREADS / 32)
#define PAIRS_PER_WAVE (NPAIRS / NWAVES)   // 512
#define CHUNKS   (PAIRS_PER_WAVE / 16)     // 32 WMMAs per wave per gate
#define GROUP    8                          // chunks batched for DS latency hiding
#define SPT      (NSTATES / THREADS)        // 32 states per thread

// Apply fused U = Rz(t3)*Ry(t2)*Rz(t1) on qubit bit-position p (0..11).
// A fragment (16x4 fp32): lane L (0-15) holds A[L][K=0..1], lane L+16 holds
// A[L][K=2..3]; we set A[M] = M4[M & 3] (4 stacked copies of the real 4x4).
// B fragment (4x16): lane n v0/v1 = (re0,im0) of pair n; lane n+16 = (re1,im1).
// D fragment: every lane's d[0..3] = (re0',im0',re1',im1') of pair (lane&15).
__device__ __forceinline__ void apply_fused_u(float2* __restrict__ S,
                                              float t1, float t2, float t3,
                                              int p, int lane, int wave) {
    const float alpha = 0.5f * (t1 + t3);
    const float beta  = 0.5f * (t1 - t3);
    const float htheta = 0.5f * t2;
    const float ca = cosf(alpha), sa = sinf(alpha);
    const float cb = cosf(beta),  sb = sinf(beta);
    const float c2 = cosf(htheta), s2 = sinf(htheta);

    // U00 = c2*e^{-ia}; U01 = -s2*e^{+ib}; U10 = s2*e^{-ib}; U11 = c2*e^{+ia}
    const float2 U00 = make_float2( c2 * ca, -c2 * sa);
    const float2 U01 = make_float2(-s2 * cb, -s2 * sb);
    const float2 U10 = make_float2( s2 * cb, -s2 * sb);
    const float2 U11 = make_float2( c2 * ca,  c2 * sa);

    const int g  = lane & 3;        // row of M4 this lane supplies
    const int hi = lane >> 4;       // 0: columns {0,1}, 1: columns {2,3}
    float2 u;
    if ((g >> 1) == 0) u = hi ? U01 : U00;
    else               u = hi ? U11 : U10;
    // even rows (re outputs): [ur, -ui]; odd rows (im outputs): [ui, ur]
    v2f a;
    a.x = (g & 1) ? u.y :  u.x;
    a.y = (g & 1) ? u.x : -u.y;

    const int      col  = lane & 15;
    const unsigned mask = (1u << p) - 1u;
    const int      base = wave * PAIRS_PER_WAVE;
    const unsigned hbit = ((unsigned)hi) << p;

    for (int cg = 0; cg < CHUNKS; cg += GROUP) {
        unsigned addr[GROUP];
        float2   vin[GROUP];
        // Batch GROUP ds_load_b64's so they are all in flight together.
#pragma unroll
        for (int j = 0; j < GROUP; ++j) {
            const int      pn   = base + (cg + j) * 16 + col;
            const unsigned idx0 = (((unsigned)pn & ~mask) << 1) | ((unsigned)pn & mask);
            addr[j] = idx0 | hbit;             // lane<16 -> amp0, lane>=16 -> amp1
            vin[j]  = S[addr[j]];              // ds_load_b64
        }
#pragma unroll
        for (int j = 0; j < GROUP; ++j) {
            v2f b; b.x = vin[j].x; b.y = vin[j].y;
            v8f cz = {};
            v8f d = __builtin_amdgcn_wmma_f32_16x16x4_f32(
                /*neg_a=*/false, a, /*neg_b=*/false, b,
                /*c_mod=*/(short)0, cz, /*reuse_a=*/false, /*reuse_b=*/false);
            float2 vout;
            vout.x = hi ? d[2] : d[0];
            vout.y = hi ? d[3] : d[1];
            S[addr[j]] = vout;                 // ds_store_b64
        }
    }
}

// One layer's full CNOT block as a single permutation.
// Gate i (applied i = 0..11 in order): control qubit i, target (i+h+1)%12.
// psi_final[x] = psi0[G_{i=0}(G_{i=1}(...(G_{i=11}(x))...))]  (CNOT self-inverse)
__device__ __forceinline__ void apply_cnot_block(float2* __restrict__ S,
                                                 int h, int tid) {
    float2 vals[SPT];
#pragma unroll
    for (int k = 0; k < SPT; ++k) {
        unsigned y = (unsigned)(tid + k * THREADS);
#pragma unroll
        for (int i = NQ - 1; i >= 0; --i) {    // reverse gate order on the index
            const int pc = 11 - i;
            const int pt = 11 - ((i + h + 1) % NQ);
            y ^= ((y >> pc) & 1u) << pt;
        }
        vals[k] = S[y];                        // gather
    }
    __syncthreads();                           // all gathers done before scatter
#pragma unroll
    for (int k = 0; k < SPT; ++k) {
        S[tid + k * THREADS] = vals[k];
    }
}

__global__ void __launch_bounds__(THREADS)
qcircuit_kernel(const float* __restrict__ x,   // (nCirc, 12) angles
                const float* __restrict__ w,   // (3, 12, 3) weights
                float* __restrict__ out) {     // (nCirc, 12) <Z_i>
    __shared__ float2 S[NSTATES];              // 32 KB state
    __shared__ float  ang[NQ];
    __shared__ float  wsh[NLAY * NQ * 3];
    __shared__ float  zacc[NQ];

    const int tid  = threadIdx.x;
    const int lane = tid & 31;
    const int wave = tid >> 5;
    const int blk  = blockIdx.x;

    if (tid < NQ) { ang[tid] = x[blk * NQ + tid]; zacc[tid] = 0.0f; }
    for (int i = tid; i < NLAY * NQ * 3; i += THREADS) wsh[i] = w[i];

    // |0...0>
    for (int s = tid; s < NSTATES; s += THREADS) S[s] = make_float2(0.0f, 0.0f);
    __syncthreads();
    if (tid == 0) S[0] = make_float2(1.0f, 0.0f);
    __syncthreads();

    // Angle embedding: Rz(a) Ry(a) Rz(a) per qubit (qubit n -> bit 11-n).
#pragma unroll 1
    for (int n = 0; n < NQ; ++n) {
        const float aq = ang[n];
        apply_fused_u(S, aq, aq, aq, 11 - n, lane, wave);
        __syncthreads();
    }

    // Strongly entangling layers.
#pragma unroll 1
    for (int h = 0; h < NLAY; ++h) {
#pragma unroll 1
        for (int i = 0; i < NQ; ++i) {
            const int o = (h * NQ + i) * 3;
            apply_fused_u(S, wsh[o + 0], wsh[o + 1], wsh[o + 2], 11 - i, lane, wave);
            __syncthreads();
        }
        apply_cnot_block(S, h, tid);
        __syncthreads();
    }

    // <Z_i> = sum_s |psi_s|^2 * (1 - 2*bit_i(s))
    float z[NQ];
#pragma unroll
    for (int i = 0; i < NQ; ++i) z[i] = 0.0f;
    for (int s = tid; s < NSTATES; s += THREADS) {
        const float2 v = S[s];
        const float pr = v.x * v.x + v.y * v.y;
#pragma unroll
        for (int i = 0; i < NQ; ++i)
            z[i] += ((s >> (11 - i)) & 1) ? -pr : pr;
    }
#pragma unroll
    for (int i = 0; i < NQ; ++i) atomicAdd(&zacc[i], z[i]);  // ds_add_f32
    __syncthreads();
    if (tid < NQ) out[blk * NQ + tid] = zacc[tid];
}

extern "C" void kernel_launch(void* const* d_in, const int* in_sizes, int n_in,
                              void* d_out, int out_size, void* d_ws, size_t ws_size,
                              hipStream_t stream) {
    (void)n_in; (void)out_size; (void)d_ws; (void)ws_size;
    const float* x = (const float*)d_in[0];   // (8,32,12) float32
    const float* w = (const float*)d_in[1];   // (3,12,3) float32
    float* out = (float*)d_out;               // (8,32,12) float32
    const int nCirc = in_sizes[0] / NQ;       // 256 circuits
    qcircuit_kernel<<<nCirc, THREADS, 0, stream>>>(x, w, out);
}